// GCN_25752623907390
// MI455X (gfx1250) — compile-verified
//
#include <hip/hip_runtime.h>

// GCN on MI455X (gfx1250). Aggregation (edge scatter-add) dominates and is
// L2-resident (51.2MB features << 192MB L2); GEMMs use fp32 WMMA 16x16x4.

#define N_NODES 100000
#define N_EDGES 1600000

typedef float v2f __attribute__((ext_vector_type(2)));
typedef float v8f __attribute__((ext_vector_type(8)));

// ---------------------------------------------------------------- utilities
__global__ void zero_f32(float* __restrict__ p, long n) {
  long i = (long)blockIdx.x * blockDim.x + threadIdx.x;
  long stride = (long)gridDim.x * blockDim.x;
  for (; i < n; i += stride) p[i] = 0.0f;
}

__global__ void deg_kernel(const int* __restrict__ src, const int* __restrict__ dst,
                           float* __restrict__ deg_s, float* __restrict__ deg_d, int E) {
  int i = blockIdx.x * blockDim.x + threadIdx.x;
  if (i < E) {
    atomicAdd(&deg_s[src[i]], 1.0f);
    atomicAdd(&deg_d[dst[i]], 1.0f);
  }
}

// d[i] = rsqrt(max(deg,1))  (applied to both deg arrays, which are contiguous)
__global__ void norm_kernel(float* __restrict__ d, int n) {
  int i = blockIdx.x * blockDim.x + threadIdx.x;
  if (i < n) d[i] = rsqrtf(fmaxf(d[i], 1.0f));
}

// ------------------------------------------------ GEMM: X = (H * ns) @ W
// One wave (32 threads) computes one 16x16 output tile with
// V_WMMA_F32_16X16X4_F32, K-loop in steps of 4. EXEC stays all-ones.
__global__ __launch_bounds__(32)
void gemm_wmma(const float* __restrict__ H, const float* __restrict__ ns,
               const float* __restrict__ W, float* __restrict__ X,
               int in, int out) {
  const int lane = threadIdx.x;       // 0..31
  const int lo   = lane & 15;
  const int hi   = lane >> 4;         // 0 or 1
  const int row  = blockIdx.x * 16 + lo;   // node index (A rows / D rows base)
  const int col  = blockIdx.y * 16 + lo;   // output feature (B/D cols)
  const float s  = ns[row];
  const float* __restrict__ hrow = H + (long)row * in;

  v8f acc = {};
  for (int k = 0; k < in; k += 4) {
    const int kk = k + 2 * hi;        // lanes 0-15: K=k,k+1 ; lanes 16-31: K=k+2,k+3
    v2f a, b;
    a[0] = hrow[kk]     * s;
    a[1] = hrow[kk + 1] * s;
    b[0] = W[(long)kk * out + col];
    b[1] = W[(long)(kk + 1) * out + col];
    acc = __builtin_amdgcn_wmma_f32_16x16x4_f32(
        /*neg_a=*/false, a, /*neg_b=*/false, b,
        /*c_mod=*/(short)0, acc, /*reuse_a=*/false, /*reuse_b=*/false);
  }

  const int mbase = blockIdx.x * 16 + 8 * hi;  // c[j] -> M = j + 8*hi
#pragma unroll
  for (int j = 0; j < 8; ++j)
    X[(long)(mbase + j) * out + col] = acc[j];
}

// Scalar GEMM for the final 64->2 layer (doesn't tile to 16x16).
__global__ void gemm_small(const float* __restrict__ H, const float* __restrict__ ns,
                           const float* __restrict__ W, float* __restrict__ X,
                           int n, int in, int out) {
  int i = blockIdx.x * blockDim.x + threadIdx.x;
  if (i >= n) return;
  const float s = ns[i];
  const float* __restrict__ hrow = H + (long)i * in;
  for (int c = 0; c < out; ++c) {
    float acc = 0.0f;
    for (int k = 0; k < in; ++k) acc = fmaf(hrow[k] * s, W[(long)k * out + c], acc);
    X[(long)i * out + c] = acc;
  }
}

// ------------------------------------------------ scatter: agg[dst] += X[src]
// One float4 chunk per thread; 32 (out=128) or 16 (out=64) consecutive threads cover one edge's row,
// so atomics hit consecutive addresses in one L2 line (good atomic coalescing).
__global__ void scatter4(const float* __restrict__ X, const int* __restrict__ src,
                         const int* __restrict__ dst, float* __restrict__ AGG,
                         long total, int out, int shift) {
  long idx = (long)blockIdx.x * blockDim.x + threadIdx.x;
  if (idx >= total) return;
  const int e = (int)(idx >> shift);
  const int c = (int)(idx & ((1 << shift) - 1));
  const int s = src[e], d = dst[e];
  const float4 v = ((const float4*)(X + (long)s * out))[c];
  float* dp = AGG + (long)d * out + 4 * c;
  atomicAdd(dp + 0, v.x);
  atomicAdd(dp + 1, v.y);
  atomicAdd(dp + 2, v.z);
  atomicAdd(dp + 3, v.w);
}

__global__ void scatter2(const float* __restrict__ X, const int* __restrict__ src,
                         const int* __restrict__ dst, float* __restrict__ AGG, int E) {
  int e = blockIdx.x * blockDim.x + threadIdx.x;
  if (e >= E) return;
  const float2 v = *(const float2*)(X + 2L * src[e]);
  float* dp = AGG + 2L * dst[e];
  atomicAdd(dp + 0, v.x);
  atomicAdd(dp + 1, v.y);
}

// --------------------------------- epilogue: h = relu(agg*norm_dst + gb + sb)
__global__ void epilogue(float* __restrict__ A, const float* __restrict__ nd,
                         const float* __restrict__ gb, const float* __restrict__ sb,
                         long total, int oshift) {
  long i = (long)blockIdx.x * blockDim.x + threadIdx.x;
  if (i >= total) return;
  const int node = (int)(i >> oshift);
  const int f    = (int)(i & ((1 << oshift) - 1));
  const float v = A[i] * nd[node] + gb[f] + sb[0];
  A[i] = fmaxf(v, 0.0f);
}

__global__ void softmax2(const float* __restrict__ H, float* __restrict__ O, int n) {
  int i = blockIdx.x * blockDim.x + threadIdx.x;
  if (i >= n) return;
  const float a = H[2L * i], b = H[2L * i + 1];
  const float m = fmaxf(a, b);
  const float ea = expf(a - m), eb = expf(b - m);
  const float inv = 1.0f / (ea + eb);
  O[2L * i]     = ea * inv;
  O[2L * i + 1] = eb * inv;
}

// ---------------------------------------------------------------- launcher
extern "C" void kernel_launch(void* const* d_in, const int* in_sizes, int n_in,
                              void* d_out, int out_size, void* d_ws, size_t ws_size,
                              hipStream_t stream) {
  const float* feats = (const float*)d_in[0];
  const int*   src   = (const int*)d_in[1];
  const int*   dst   = (const int*)d_in[2];
  const float* W[10]; const float* GB[10]; const float* SB[10];
  for (int l = 0; l < 10; ++l) {
    W[l]  = (const float*)d_in[3 + l];
    GB[l] = (const float*)d_in[13 + l];
    SB[l] = (const float*)d_in[23 + l];
  }
  static const int IN[10]  = {128,128,128,128,128,128,128, 64, 64, 64};
  static const int OUT[10] = {128,128,128,128,128,128, 64, 64, 64,  2};

  float* ws     = (float*)d_ws;
  float* norm_s = ws;                                  // [N]
  float* norm_d = ws + N_NODES;                        // [N] (contiguous with norm_s)
  float* Xb     = ws + 2L * N_NODES;                   // [N,128]
  float* bufA   = Xb   + (long)N_NODES * 128;          // [N,128]
  float* bufB   = bufA + (long)N_NODES * 128;          // [N,128]

  // degrees -> symmetric norms (recomputed every call: ws is not trusted)
  zero_f32<<<2048, 256, 0, stream>>>(norm_s, 2L * N_NODES);
  deg_kernel<<<(N_EDGES + 255) / 256, 256, 0, stream>>>(src, dst, norm_s, norm_d, N_EDGES);
  norm_kernel<<<(2 * N_NODES + 255) / 256, 256, 0, stream>>>(norm_s, 2 * N_NODES);

  const float* Hcur = feats;
  float* pingpong[2] = {bufA, bufB};

  for (int l = 0; l < 10; ++l) {
    const int in = IN[l], out = OUT[l];
    float* AGG = pingpong[l & 1];

    // 1) X = (Hcur * norm_s) @ W[l]
    if (out >= 16) {
      dim3 g(N_NODES / 16, out / 16);   // 100000/16 = 6250 exactly
      gemm_wmma<<<g, 32, 0, stream>>>(Hcur, norm_s, W[l], Xb, in, out);
    } else {
      gemm_small<<<(N_NODES + 255) / 256, 256, 0, stream>>>(Hcur, norm_s, W[l], Xb,
                                                            N_NODES, in, out);
    }

    // 2) agg = segment_sum(X[src], dst)
    zero_f32<<<4096, 256, 0, stream>>>(AGG, (long)N_NODES * out);
    if (out >= 4) {
      const int shift = (out == 128) ? 5 : 4;          // chunks of 4 floats
      const long total = (long)N_EDGES * (out >> 2);
      scatter4<<<(int)((total + 255) / 256), 256, 0, stream>>>(Xb, src, dst, AGG,
                                                               total, out, shift);
    } else {
      scatter2<<<(N_EDGES + 255) / 256, 256, 0, stream>>>(Xb, src, dst, AGG, N_EDGES);
    }

    // 3) h = relu(agg * norm_d + gb + sb)
    const int oshift = (out == 128) ? 7 : (out == 64) ? 6 : 1;
    const long total = (long)N_NODES * out;
    epilogue<<<(int)((total + 255) / 256), 256, 0, stream>>>(AGG, norm_d, GB[l], SB[l],
                                                             total, oshift);
    Hcur = AGG;
  }

  softmax2<<<(N_NODES + 255) / 256, 256, 0, stream>>>(Hcur, (float*)d_out, N_NODES);
}